// GAT_AD_55817394978970
// MI455X (gfx1250) — compile-verified
//
#include <hip/hip_runtime.h>
#include <hip/hip_bf16.h>

typedef float v2f __attribute__((ext_vector_type(2)));
typedef float v8f __attribute__((ext_vector_type(8)));

#define WDIM  64      // x feature dim (window)
#define HDIM  128     // hidden dim
#define TCOLS 256     // combined node feature: [x@w1_top + b1 | x@w1_bot]

// ---------------------------------------------------------------------------
// Kernel 0: init per-node buffers (mkey=0 decodes below any real logit).
// ---------------------------------------------------------------------------
__global__ void init_node_bufs(unsigned* __restrict__ mkey,
                               float* __restrict__ denom,
                               float* __restrict__ yhat, int n)
{
    int i = blockIdx.x * blockDim.x + threadIdx.x;
    if (i < n) { mkey[i] = 0u; denom[i] = 0.0f; yhat[i] = 0.0f; }
}

// ---------------------------------------------------------------------------
// Kernel 1: T[m, 0:128]   = x[m] @ w1[0:64, :]   + b1
//           T[m, 128:256] = x[m] @ w1[64:128, :]
// One wave per 16-node tile, V_WMMA_F32_16X16X4_F32, K=64 (16 chunks),
// 16 N-tiles of 16 cols. Combined 64x256 weight + expanded bias in LDS.
// ---------------------------------------------------------------------------
__global__ void __launch_bounds__(256)
node_mlp_wmma(const float* __restrict__ x, const float* __restrict__ w1,
              const float* __restrict__ b1, float* __restrict__ T, int nNodes)
{
    __shared__ float sW[WDIM * TCOLS];   // 64 x 256 f32 = 64 KB
    __shared__ float sBias[TCOLS];       // [b1 | 0], branch-free bias fetch
    for (int i = threadIdx.x; i < WDIM * TCOLS; i += 256) {
        int k = i >> 8;       // 0..63
        int j = i & 255;      // 0..255
        sW[i] = (j < HDIM) ? w1[k * HDIM + j]
                           : w1[(WDIM + k) * HDIM + (j - HDIM)];
    }
    sBias[threadIdx.x] = (threadIdx.x < HDIM) ? b1[threadIdx.x] : 0.0f;
    __syncthreads();

    const int wave  = threadIdx.x >> 5;
    const int lane  = threadIdx.x & 31;
    const int nTile = (nNodes + 15) >> 4;
    const int tile  = blockIdx.x * 8 + wave;
    if (tile >= nTile) return;           // uniform per-wave: EXEC stays all-1s

    const int m0       = tile << 4;
    const bool fullTile = (m0 + 16 <= nNodes);   // uniform across wave
    const int half     = lane >> 4;      // 0: lanes 0-15, 1: lanes 16-31
    const int lm       = lane & 15;

    int rowA = m0 + lm;
    if (rowA > nNodes - 1) rowA = nNodes - 1;    // clamp (dups only if ragged)
    const float* __restrict__ xr = x + (size_t)rowA * WDIM;

    // A fragments: 16x4 f32 per chunk; VGPR0 = K {0,2}, VGPR1 = K {1,3}
    v2f a[16];
    #pragma unroll
    for (int c = 0; c < 16; ++c) {
        const int col = (c << 2) + (half << 1);
        a[c].x = xr[col];
        a[c].y = xr[col + 1];
    }

    #pragma unroll 1
    for (int t = 0; t < 16; ++t) {
        const int n = (t << 4) + lm;                 // output column
        const float bias = sBias[n];
        v8f acc;
        #pragma unroll
        for (int r = 0; r < 8; ++r) acc[r] = bias;   // fold b1 into C

        #pragma unroll
        for (int c = 0; c < 16; ++c) {
            const int kb = (c << 2) + (half << 1);   // B 4x16: VGPR0=K{0,2}, VGPR1=K{1,3}
            v2f b;
            b.x = sW[kb * TCOLS + n];
            b.y = sW[(kb + 1) * TCOLS + n];
            acc = __builtin_amdgcn_wmma_f32_16x16x4_f32(
                      false, a[c], false, b, (short)0, acc, false, false);
        }

        // C/D layout: VGPR r -> M = r + 8*half, N = n
        float* __restrict__ Tp = T + (size_t)m0 * TCOLS + n;
        if (fullTile) {                              // uniform: no exec churn
            #pragma unroll
            for (int r = 0; r < 8; ++r) {
                const int M = r + (half << 3);
                Tp[(size_t)M * TCOLS] = acc[r];
            }
        } else {                                     // ragged last tile only
            #pragma unroll
            for (int r = 0; r < 8; ++r) {
                const int M = r + (half << 3);
                if (m0 + M < nNodes) Tp[(size_t)M * TCOLS] = acc[r];
            }
        }
    }
}

// ---------------------------------------------------------------------------
// Kernel 2: per edge, logit = sum_n relu(T[src][n] + T[dst][128+n]) * w2[n] + b2
// 4 lanes per edge (each lane: 8 float4 chunks), shfl reduce, then
// segment-max via monotone uint-key atomicMax on mkey[dst].
// ---------------------------------------------------------------------------
__device__ __forceinline__ unsigned f32_to_key(float f) {
    unsigned u = __float_as_uint(f);
    return (u & 0x80000000u) ? ~u : (u | 0x80000000u);
}
__device__ __forceinline__ float key_to_f32(unsigned k) {
    unsigned u = (k & 0x80000000u) ? (k ^ 0x80000000u) : ~k;
    return __uint_as_float(u);
}

__global__ void __launch_bounds__(256)
edge_logits(const int* __restrict__ src, const int* __restrict__ dst,
            const float* __restrict__ T, const float* __restrict__ w2,
            const float* __restrict__ b2, float* __restrict__ logits,
            unsigned* __restrict__ mkey, int nEdges)
{
    const int gid = blockIdx.x * blockDim.x + threadIdx.x;
    const int e   = gid >> 2;
    const int sub = gid & 3;
    if (e >= nEdges) return;

    const int s = src[e];
    const int d = dst[e];
    const float4* __restrict__ Ps = (const float4*)(T + (size_t)s * TCOLS);
    const float4* __restrict__ Qd = (const float4*)(T + (size_t)d * TCOLS + HDIM);
    const float4* __restrict__ W2 = (const float4*)w2;

    float acc = 0.0f;
    #pragma unroll
    for (int i = 0; i < 8; ++i) {
        const int idx = sub + (i << 2);     // 32 float4 chunks, 4 lanes interleave
        float4 p = Ps[idx];
        float4 q = Qd[idx];
        float4 w = W2[idx];
        acc += fmaxf(p.x + q.x, 0.0f) * w.x;
        acc += fmaxf(p.y + q.y, 0.0f) * w.y;
        acc += fmaxf(p.z + q.z, 0.0f) * w.z;
        acc += fmaxf(p.w + q.w, 0.0f) * w.w;
    }
    acc += __shfl_xor(acc, 1, 32);
    acc += __shfl_xor(acc, 2, 32);

    if (sub == 0) {
        const float logit = acc + b2[0];    // TEMP == 1.0
        logits[e] = logit;
        atomicMax(&mkey[d], f32_to_key(logit));
    }
}

// ---------------------------------------------------------------------------
// Kernel 3: e = exp(logit - m[dst]) (in-place over logits); denom[dst] += e
// ---------------------------------------------------------------------------
__global__ void __launch_bounds__(256)
edge_exp(const int* __restrict__ dst, float* __restrict__ logits,
         const unsigned* __restrict__ mkey, float* __restrict__ denom, int nEdges)
{
    const int e = blockIdx.x * blockDim.x + threadIdx.x;
    if (e >= nEdges) return;
    const int d = dst[e];
    const float m  = key_to_f32(mkey[d]);
    const float ex = expf(logits[e] - m);
    logits[e] = ex;                         // reuse buffer for exp values
    atomicAdd(&denom[d], ex);
}

// ---------------------------------------------------------------------------
// Kernel 4: alpha = e / denom[dst]; yhat[dst] += y[src] * alpha
// ---------------------------------------------------------------------------
__global__ void __launch_bounds__(256)
edge_final(const int* __restrict__ src, const int* __restrict__ dst,
           const float* __restrict__ y, const float* __restrict__ evals,
           const float* __restrict__ denom, float* __restrict__ yhat,
           float* __restrict__ alpha, int nEdges)
{
    const int e = blockIdx.x * blockDim.x + threadIdx.x;
    if (e >= nEdges) return;
    const int d = dst[e];
    const float a = evals[e] / denom[d];
    alpha[e] = a;
    atomicAdd(&yhat[d], y[src[e]] * a);
}

// ---------------------------------------------------------------------------
extern "C" void kernel_launch(void* const* d_in, const int* in_sizes, int n_in,
                              void* d_out, int out_size, void* d_ws, size_t ws_size,
                              hipStream_t stream)
{
    const float* x    = (const float*)d_in[0];
    const float* y    = (const float*)d_in[1];
    const int*   eidx = (const int*)  d_in[2];
    const float* w1   = (const float*)d_in[3];
    const float* b1   = (const float*)d_in[4];
    const float* w2   = (const float*)d_in[5];
    const float* b2   = (const float*)d_in[6];

    const int N = in_sizes[1];          // 50000 nodes
    const int E = in_sizes[2] / 2;      // 1.6M edges
    const int* src = eidx;              // edge_index[0]
    const int* dst = eidx + E;          // edge_index[1]

    // workspace layout (floats): T[N*256] | logits/e[E] | mkey[N] | denom[N]
    float*    T      = (float*)d_ws;
    float*    logits = T + (size_t)N * TCOLS;
    unsigned* mkey   = (unsigned*)(logits + E);
    float*    denom  = (float*)(mkey + N);

    float* yhat  = (float*)d_out;       // [N]
    float* alpha = yhat + N;            // [E]

    const int nTile = (N + 15) >> 4;

    init_node_bufs<<<(N + 255) / 256, 256, 0, stream>>>(mkey, denom, yhat, N);
    node_mlp_wmma<<<(nTile + 7) / 8, 256, 0, stream>>>(x, w1, b1, T, N);
    edge_logits<<<((E * 4) + 255) / 256, 256, 0, stream>>>(src, dst, T, w2, b2,
                                                           logits, mkey, E);
    edge_exp<<<(E + 255) / 256, 256, 0, stream>>>(dst, logits, mkey, denom, E);
    edge_final<<<(E + 255) / 256, 256, 0, stream>>>(src, dst, y, logits, denom,
                                                    yhat, alpha, E);
}